// LSDQN_layer_12867722019088
// MI455X (gfx1250) — compile-verified
//
#include <hip/hip_runtime.h>
#include <hip/hip_bf16.h>

typedef float v2f __attribute__((ext_vector_type(2)));
typedef float v8f __attribute__((ext_vector_type(8)));

#define D 64  // D_IN == D_H == 64

// ---------------------------------------------------------------------------
// Kernel 0: zero the accumulators (acc1[N*64], acc2[N*64], counts[N]) — the
// harness poisons d_ws with 0xAA, and we must be re-runnable under graph replay.
// ---------------------------------------------------------------------------
__global__ void zero_ws_kernel(float* __restrict__ p, long nElem) {
    long i = (long)blockIdx.x * blockDim.x + threadIdx.x;
    if (i < nElem) p[i] = 0.0f;
}

// ---------------------------------------------------------------------------
// Kernel 1: edge scatter. One half-wave (16 lanes) per edge; each lane owns a
// contiguous float4 slice of the 64-wide feature row. The u[col] gather is a
// single coalesced 256B row read; scatter is f32 atomics into L2-resident acc.
// ---------------------------------------------------------------------------
__global__ void edge_scatter_kernel(const float* __restrict__ edge_attr,
                                    const int*   __restrict__ edge_index, // [2,E]
                                    const float* __restrict__ u,
                                    const float* __restrict__ W3,  // [64,1] flat
                                    const float* __restrict__ b3,  // [64]
                                    float* __restrict__ acc1,
                                    float* __restrict__ acc2,
                                    float* __restrict__ counts,
                                    int E) {
    long tid  = (long)blockIdx.x * blockDim.x + threadIdx.x;
    long e    = tid >> 4;                 // edge id (half-wave per edge)
    if (e >= E) return;
    int  sub  = threadIdx.x & 15;
    int  j    = sub * 4;

    float a = edge_attr[e];
    int   r = edge_index[e];              // row (destination / segment id)
    int   c = edge_index[(long)E + e];    // col (source of u)

    const float4 uv = *(const float4*)(u  + (long)c * D + j);
    const float4 w  = *(const float4*)(W3 + j);
    const float4 bb = *(const float4*)(b3 + j);

    float* p1 = acc1 + (long)r * D + j;
    float* p2 = acc2 + (long)r * D + j;

    atomicAdd(p1 + 0, a * uv.x);
    atomicAdd(p1 + 1, a * uv.y);
    atomicAdd(p1 + 2, a * uv.z);
    atomicAdd(p1 + 3, a * uv.w);

    atomicAdd(p2 + 0, fmaxf(a * w.x + bb.x, 0.0f));
    atomicAdd(p2 + 1, fmaxf(a * w.y + bb.y, 0.0f));
    atomicAdd(p2 + 2, fmaxf(a * w.z + bb.z, 0.0f));
    atomicAdd(p2 + 3, fmaxf(a * w.w + bb.w, 0.0f));

    if (sub == 0) atomicAdd(counts + r, 1.0f);
}

// ---------------------------------------------------------------------------
// Kernel 2: node pass via V_WMMA_F32_16X16X4_F32.
// One wave computes a 16-node x 64-output tile:
//   out = relu( X@W0^T + (acc1/deg)@W1^T + (acc2/deg)@W2^T + b0+b1+b2 )
// A layout (16x4 f32): lanes 0-15 carry M=0..15; VGPR0/1 = K pair selected by
// lane half  ->  per-lane contiguous float2 at [m*64 + kb].
// B layout (4x16): lane carries column n; VGPR j / lane half select K
//   ->  B[k][n] = W[n][k] is a per-lane contiguous float2 at [n*64 + kb].
// C/D layout: component i = row m0+i(+8*half), column n = nt*16 + (lane&15).
// ---------------------------------------------------------------------------
__global__ void node_wmma_kernel(const float* __restrict__ x,
                                 const float* __restrict__ acc1,
                                 const float* __restrict__ acc2,
                                 const float* __restrict__ counts,
                                 const float* __restrict__ W0, const float* __restrict__ b0,
                                 const float* __restrict__ W1, const float* __restrict__ b1,
                                 const float* __restrict__ W2, const float* __restrict__ b2,
                                 float* __restrict__ out, int n) {
    const int lane = threadIdx.x & 31;
    const int wib  = threadIdx.x >> 5;
    const int m0   = (blockIdx.x * (blockDim.x >> 5) + wib) * 16;
    if (m0 >= n) return;                  // wave-uniform: EXEC stays all-1s for WMMA

    const int ml = lane & 15;             // row-in-tile for A; column for B/C/D
    const int kh = lane >> 4;             // K-half selector
    const int m  = m0 + ml;               // this lane's node row (A side)

    const float inv = 1.0f / fmaxf(counts[m], 1.0f);

    const float* __restrict__ xrow  = x    + (long)m * D;
    const float* __restrict__ a1row = acc1 + (long)m * D;
    const float* __restrict__ a2row = acc2 + (long)m * D;

    v8f acc[4];
#pragma unroll
    for (int nt = 0; nt < 4; ++nt) {
        const int ncol = nt * 16 + ml;
        const float bs = b0[ncol] + b1[ncol] + b2[ncol];
        acc[nt] = (v8f){bs, bs, bs, bs, bs, bs, bs, bs};
    }

#pragma unroll
    for (int nt = 0; nt < 4; ++nt) {
        const int ncol = nt * 16 + ml;
        const float* __restrict__ w0row = W0 + (long)ncol * D;
        const float* __restrict__ w1row = W1 + (long)ncol * D;
        const float* __restrict__ w2row = W2 + (long)ncol * D;

        // term 1: X @ W0^T
#pragma unroll
        for (int kk = 0; kk < 16; ++kk) {
            const int kb = kk * 4 + 2 * kh;
            v2f a = *(const v2f*)(xrow + kb);
            v2f b = *(const v2f*)(w0row + kb);
            acc[nt] = __builtin_amdgcn_wmma_f32_16x16x4_f32(
                false, a, false, b, (short)0, acc[nt], false, false);
        }
        // term 2: (acc1/deg) @ W1^T  — fold 1/deg into the A fragment
#pragma unroll
        for (int kk = 0; kk < 16; ++kk) {
            const int kb = kk * 4 + 2 * kh;
            v2f a = *(const v2f*)(a1row + kb);
            a.x *= inv; a.y *= inv;
            v2f b = *(const v2f*)(w1row + kb);
            acc[nt] = __builtin_amdgcn_wmma_f32_16x16x4_f32(
                false, a, false, b, (short)0, acc[nt], false, false);
        }
        // term 3: (acc2/deg) @ W2^T
#pragma unroll
        for (int kk = 0; kk < 16; ++kk) {
            const int kb = kk * 4 + 2 * kh;
            v2f a = *(const v2f*)(a2row + kb);
            a.x *= inv; a.y *= inv;
            v2f b = *(const v2f*)(w2row + kb);
            acc[nt] = __builtin_amdgcn_wmma_f32_16x16x4_f32(
                false, a, false, b, (short)0, acc[nt], false, false);
        }
    }

    // ReLU + store: component i -> row m0 + i + 8*kh, column nt*16 + ml
#pragma unroll
    for (int nt = 0; nt < 4; ++nt) {
        const int ncol = nt * 16 + ml;
#pragma unroll
        for (int i = 0; i < 8; ++i) {
            const int row = m0 + i + 8 * kh;
            out[(long)row * D + ncol] = fmaxf(acc[nt][i], 0.0f);
        }
    }
}

// ---------------------------------------------------------------------------
extern "C" void kernel_launch(void* const* d_in, const int* in_sizes, int n_in,
                              void* d_out, int out_size, void* d_ws, size_t ws_size,
                              hipStream_t stream) {
    (void)n_in; (void)out_size; (void)ws_size;

    const float* x          = (const float*)d_in[0];
    const float* edge_attr  = (const float*)d_in[1];
    const float* u          = (const float*)d_in[2];
    const float* W0         = (const float*)d_in[3];
    const float* b0         = (const float*)d_in[4];
    const float* W1         = (const float*)d_in[5];
    const float* b1         = (const float*)d_in[6];
    const float* W2         = (const float*)d_in[7];
    const float* b2         = (const float*)d_in[8];
    const float* W3         = (const float*)d_in[9];
    const float* b3         = (const float*)d_in[10];
    const int*   edge_index = (const int*)d_in[11];

    const int N = in_sizes[0] / D;   // 100000
    const int E = in_sizes[1];       // 1600000 (edge_attr is [E,1])

    float* acc1   = (float*)d_ws;
    float* acc2   = acc1 + (size_t)N * D;
    float* counts = acc2 + (size_t)N * D;

    // 0) zero accumulators + counts
    {
        long tot = (long)N * D * 2 + N;
        int blocks = (int)((tot + 255) / 256);
        zero_ws_kernel<<<blocks, 256, 0, stream>>>((float*)d_ws, tot);
    }
    // 1) edge scatter (half-wave per edge)
    {
        long threads = (long)E * 16;
        int blocks = (int)((threads + 255) / 256);
        edge_scatter_kernel<<<blocks, 256, 0, stream>>>(
            edge_attr, edge_index, u, W3, b3, acc1, acc2, counts, E);
    }
    // 2) node GEMMs via f32 WMMA + ReLU (one wave per 16 nodes, 8 waves/block)
    {
        int tiles  = (N + 15) / 16;
        int blocks = (tiles + 7) / 8;
        node_wmma_kernel<<<blocks, 256, 0, stream>>>(
            x, acc1, acc2, counts, W0, b0, W1, b1, W2, b2, (float*)d_out, N);
    }
}